// SelectiveSSM_35974646071953
// MI455X (gfx1250) — compile-verified
//
#include <hip/hip_runtime.h>
#include <hip/hip_bf16.h>
#include <stdint.h>

// ---------------------------------------------------------------------------
// Selective SSM (Mamba block) forward for MI455X / gfx1250.
//   B=2, L=1024, D_MODEL=2048, D_STATE=16, D_CONV=4, DT_RANK=128
// conv -> WMMA fp32 GEMM (xproj) -> WMMA fp32 GEMM+softplus (dtproj) -> scan.
// GEMMs: V_WMMA_F32_16X16X4_F32 fed from LDS tiles filled by
//        GLOBAL_LOAD_ASYNC_TO_LDS_B128 (double buffered, ASYNCcnt).
// Scan:  B/C tiles staged by the Tensor Data Mover (TENSOR_LOAD_TO_LDS,
//        TENSORcnt) while the v_exp_f32 recurrence runs.
// ---------------------------------------------------------------------------

#define D_MODEL 2048
#define D_STATE 16
#define DT_RANK 128
#define NB      2
#define NL      1024
#define NE      (DT_RANK + 2 * D_STATE)   // 160

typedef float v2f __attribute__((ext_vector_type(2)));
typedef float v8f __attribute__((ext_vector_type(8)));
typedef unsigned int v4u __attribute__((ext_vector_type(4)));
typedef int v4i __attribute__((ext_vector_type(4)));
typedef int v8i __attribute__((ext_vector_type(8)));

typedef __attribute__((address_space(1))) v4i* gv4i_ptr;   // global int4*
typedef __attribute__((address_space(3))) v4i* lv4i_ptr;   // LDS int4*

#if __has_builtin(__builtin_amdgcn_global_load_async_to_lds_b128)
#define HAVE_ASYNC 1
#else
#define HAVE_ASYNC 0
#endif

#if __has_builtin(__builtin_amdgcn_tensor_load_to_lds) && \
    __has_builtin(__builtin_amdgcn_s_wait_tensorcnt)
#define HAVE_TDM 1
#else
#define HAVE_TDM 0
#endif

#if __has_builtin(__builtin_amdgcn_s_wait_asynccnt)
#define WAIT_ASYNC(N) __builtin_amdgcn_s_wait_asynccnt(N)
#else
#define WAIT_ASYNC(N) asm volatile("s_wait_asynccnt %0" ::"i"(N) : "memory")
#endif

#define WAIT_TENSOR(N) __builtin_amdgcn_s_wait_tensorcnt(N)

// ---------------------------------------------------------------------------
// Kernel 1: depthwise causal conv1d (k=4), x:(B,L,D) -> u:(B,L,D)
// ---------------------------------------------------------------------------
__global__ __launch_bounds__(256) void conv_kernel(const float* __restrict__ x,
                                                   const float* __restrict__ cw,
                                                   const float* __restrict__ cb,
                                                   float* __restrict__ u)
{
    const int idx = blockIdx.x * 256 + threadIdx.x;   // flat (b,l,d)
    const int d = idx & (D_MODEL - 1);
    const int l = (idx >> 11) & (NL - 1);
    const int b = idx >> 21;
    const float4 w = *(const float4*)(cw + 4 * d);    // conv_w[d,0,0..3]
    const float* xb = x + (size_t)b * NL * D_MODEL + d;
    float acc = cb[d];
    if (l >= 3) {
        acc += xb[(size_t)(l - 3) * D_MODEL] * w.x;
        acc += xb[(size_t)(l - 2) * D_MODEL] * w.y;
        acc += xb[(size_t)(l - 1) * D_MODEL] * w.z;
        acc += xb[(size_t)l       * D_MODEL] * w.w;
    } else {
        if (l >= 2) acc += xb[(size_t)(l - 2) * D_MODEL] * w.y;
        if (l >= 1) acc += xb[(size_t)(l - 1) * D_MODEL] * w.z;
        acc += xb[(size_t)l * D_MODEL] * w.w;
    }
    u[idx] = acc;
}

// ---------------------------------------------------------------------------
// WMMA tile helpers. One 16x16 fp32 tile: C = A[16xK] * W[16xK]^T.
// A frag (16x4): lane holds A[lane%16][2*(lane/16)+{0,1}]
// B frag (4x16): lane holds B[2*(lane/16)+{0,1}][lane%16]  (W row-major (out,in))
// ---------------------------------------------------------------------------
#define PAD   28                 // floats per LDS row: 16B-aligned rows, all 32
                                 // lanes of the b64 frag reads hit distinct banks
#define BUFSZ (16 * PAD)         // 448 floats per buffer
#define KSTEP 16                 // k-slice per async stage

#if HAVE_ASYNC
__device__ inline void async_copy16(const float* g, float* l)
{
    // builtin signature (probed): (int4 AS1* src, int4 AS3* dst, imm, imm)
    __builtin_amdgcn_global_load_async_to_lds_b128(
        (gv4i_ptr)(uintptr_t)g,          // global VA == AS1 VA
        (lv4i_ptr)(v4i*)l,               // addrspacecast generic -> AS3
        /*imm offset=*/0, /*cpol=*/0);
}
#endif

__device__ inline v8f wmma_f32_tile(const float* __restrict__ A, int lda,
                                    const float* __restrict__ W, int ldw,
                                    int K, float* sA, float* sW)
{
    const int lane = threadIdx.x & 31;
    const int r    = lane & 15;            // A row / W row (= output col)
    const int kg   = (lane >> 4) << 1;     // k group: 0 or 2
    v8f acc = {};
#if HAVE_ASYNC
    // --- double-buffered LDS staging via GLOBAL_LOAD_ASYNC_TO_LDS_B128 ---
    const int srow = lane >> 2;            // 0..7 : tile row pair (srow, srow+8)
    const int scol = (lane & 3) << 2;      // float col 0,4,8,12 (16B per lane)
    const float* gA = A + (size_t)srow * lda + scol;
    const float* gW = W + (size_t)srow * ldw + scol;
    float* lA = sA + srow * PAD + scol;
    float* lW = sW + srow * PAD + scol;
    const size_t a8 = (size_t)8 * lda, w8 = (size_t)8 * ldw;

    // prologue: stage k=[0,16) into buffer 0 (4 coalesced 512B async copies)
    async_copy16(gA,      lA);
    async_copy16(gA + a8, lA + 8 * PAD);
    async_copy16(gW,      lW);
    async_copy16(gW + w8, lW + 8 * PAD);

    int buf = 0;
    for (int k0 = 0; k0 < K; k0 += KSTEP) {
        const int nbuf = buf ^ 1;
        if (k0 + KSTEP < K) {
            const int kn = k0 + KSTEP;
            async_copy16(gA + kn,      lA + nbuf * BUFSZ);
            async_copy16(gA + kn + a8, lA + nbuf * BUFSZ + 8 * PAD);
            async_copy16(gW + kn,      lW + nbuf * BUFSZ);
            async_copy16(gW + kn + w8, lW + nbuf * BUFSZ + 8 * PAD);
            WAIT_ASYNC(4);                 // retire current buffer (in-order)
        } else {
            WAIT_ASYNC(0);
        }
        asm volatile("" ::: "memory");     // keep ds reads below the wait
        const float* cA = sA + buf * BUFSZ + r * PAD + kg;
        const float* cW = sW + buf * BUFSZ + r * PAD + kg;
#pragma unroll
        for (int kk = 0; kk < KSTEP; kk += 4) {
            v2f a = *(const v2f*)(cA + kk);
            v2f w = *(const v2f*)(cW + kk);
            acc = __builtin_amdgcn_wmma_f32_16x16x4_f32(
                false, a, false, w, (short)0, acc, false, false);
        }
        buf = nbuf;
    }
#else
    // --- fallback: direct per-lane global fragment loads (round-1 path) ---
    (void)sA; (void)sW;
    const float* arow = A + (size_t)r * lda + kg;
    const float* wrow = W + (size_t)r * ldw + kg;
    for (int k0 = 0; k0 < K; k0 += 32) {
        __builtin_prefetch(arow + k0 + 96, 0, 3);
        __builtin_prefetch(wrow + k0 + 96, 0, 3);
#pragma unroll
        for (int kk = 0; kk < 32; kk += 4) {
            v2f a = *(const v2f*)(arow + k0 + kk);
            v2f w = *(const v2f*)(wrow + k0 + kk);
            acc = __builtin_amdgcn_wmma_f32_16x16x4_f32(
                false, a, false, w, (short)0, acc, false, false);
        }
    }
#endif
    return acc;
}

// per-block LDS for the GEMMs: 8 waves * (A + W double buffers) = 57,344 B
#define GEMM_SMEM (8 * 4 * BUFSZ)

// ---------------------------------------------------------------------------
// Kernel 2: x_dbl[BL,160] = u[BL,2048] * xproj_w[160,2048]^T
// 1280 tiles, 1 tile/wave, 8 waves/block -> 160 blocks (EXEC all-ones).
// ---------------------------------------------------------------------------
__global__ __launch_bounds__(256) void xproj_kernel(const float* __restrict__ u,
                                                    const float* __restrict__ W,
                                                    float* __restrict__ xdbl)
{
    __shared__ float smem[GEMM_SMEM];
    const int wv = threadIdx.x >> 5;
    float* sA = smem + wv * 4 * BUFSZ;
    float* sW = sA + 2 * BUFSZ;

    const int gw = blockIdx.x * 8 + wv;
    const int te = gw % (NE / 16);         // 0..9
    const int tm = gw / (NE / 16);         // 0..127
    v8f acc = wmma_f32_tile(u + (size_t)tm * 16 * D_MODEL, D_MODEL,
                            W + (size_t)te * 16 * D_MODEL, D_MODEL, D_MODEL,
                            sA, sW);
    const int lane  = threadIdx.x & 31;
    const int col   = te * 16 + (lane & 15);
    const int mrow0 = tm * 16 + ((lane >> 4) << 3);
#pragma unroll
    for (int v = 0; v < 8; ++v)
        xdbl[(size_t)(mrow0 + v) * NE + col] = acc[v];
}

// ---------------------------------------------------------------------------
// Kernel 3: delta[BL,2048] = softplus(dt[BL,128] * dtproj_w[2048,128]^T + b)
// dt = first 128 cols of x_dbl (lda=160). 16384 tiles -> 2048 blocks.
// ---------------------------------------------------------------------------
__global__ __launch_bounds__(256) void dtproj_kernel(const float* __restrict__ xdbl,
                                                     const float* __restrict__ Wdt,
                                                     const float* __restrict__ bdt,
                                                     float* __restrict__ delta)
{
    __shared__ float smem[GEMM_SMEM];
    const int wv = threadIdx.x >> 5;
    float* sA = smem + wv * 4 * BUFSZ;
    float* sW = sA + 2 * BUFSZ;

    const int gw = blockIdx.x * 8 + wv;
    const int tn = gw % (D_MODEL / 16);    // output d tile
    const int tm = gw / (D_MODEL / 16);    // row tile
    v8f acc = wmma_f32_tile(xdbl + (size_t)tm * 16 * NE, NE,
                            Wdt  + (size_t)tn * 16 * DT_RANK, DT_RANK, DT_RANK,
                            sA, sW);
    const int lane  = threadIdx.x & 31;
    const int d     = tn * 16 + (lane & 15);
    const int mrow0 = tm * 16 + ((lane >> 4) << 3);
    const float bias = bdt[d];
#pragma unroll
    for (int v = 0; v < 8; ++v) {
        float t  = acc[v] + bias;
        float sp = (t > 20.f) ? t : log1pf(__expf(t));   // softplus
        delta[(size_t)(mrow0 + v) * D_MODEL + d] = sp;
    }
}

// ---------------------------------------------------------------------------
// Kernel 4: sequential selective scan; B/C tiles staged by the TDM.
// ---------------------------------------------------------------------------
#define LCHUNK 32

#if HAVE_TDM
// TENSOR_LOAD_TO_LDS of a 2D 32x32 fp32 tile with row stride NE (=160).
// Descriptor bitfields per ISA 8.3/8.4 (2D: groups 2/3 zero, VADDR2/3 unused).
__device__ inline void tdm_load_tile32x32(const float* gsrc, float* ldst)
{
    const uint64_t ga = (uint64_t)(uintptr_t)gsrc;
    const unsigned lofs =
        (unsigned)(size_t)(__attribute__((address_space(3))) char*)ldst;
    v4u g0 = { 1u,                                          // count=1
               lofs,                                        // lds_addr
               (unsigned)(ga & 0xFFFFFFFFu),                // global_addr[31:0]
               (unsigned)(((ga >> 32) & 0x01FFFFFFu) |      // global_addr[56:32]
                          (2u << 30)) };                    // type=2 (image)
    v8i g1 = { (int)(2u << 16),      // wg_mask=0, data_size=2 (4B), no flags
               (int)(32u << 16),     // tensor_dim0 = 32 (cols 128..159)
               (int)(32u << 16),     // tensor_dim1 = 32
               (int)(32u << 16),     // tile_dim0   = 32
               (int)32,              // tile_dim1   = 32, tile_dim2 = 0
               (int)NE,              // tensor_dim0_stride = 160
               0, 0 };               // tensor_dim1_stride = 0 (2D)
    v4i gz = { 0, 0, 0, 0 };
#if defined(__clang_major__) && (__clang_major__ >= 23)
    v8i gz8 = { 0, 0, 0, 0, 0, 0, 0, 0 };
    __builtin_amdgcn_tensor_load_to_lds(g0, g1, gz, gz, gz8, 0);
#else
    __builtin_amdgcn_tensor_load_to_lds(g0, g1, gz, gz, 0);
#endif
}
#else
__device__ inline void stage_bc_coop(const float* __restrict__ src,
                                     float* __restrict__ dst, int tid)
{
#pragma unroll
    for (int i = 0; i < (LCHUNK * 32) / 256; ++i) {
        int idx = tid + i * 256;
        dst[idx] = src[(size_t)(idx >> 5) * NE + (idx & 31)];
    }
}
#endif

__global__ __launch_bounds__(256) void scan_kernel(const float* __restrict__ u,
                                                   const float* __restrict__ xdbl,
                                                   const float* __restrict__ delta,
                                                   const float* __restrict__ A_log,
                                                   const float* __restrict__ Dskip,
                                                   float* __restrict__ y)
{
    __shared__ float sBC[2][LCHUNK][2 * D_STATE];  // [buf][l][0..15]=B, [16..31]=C
    const int tid = threadIdx.x;
    const int b   = blockIdx.x >> 3;
    const int d   = ((blockIdx.x & 7) << 8) + tid;

    float a[D_STATE], h[D_STATE];
#pragma unroll
    for (int n = 0; n < D_STATE; ++n) {
        a[n] = -__expf(A_log[(size_t)d * D_STATE + n]);   // A = -exp(A_log)
        h[n] = 0.f;
    }
    const float dsk   = Dskip[d];
    const size_t rowb = (size_t)b * NL;
    const float* bcsrc = xdbl + rowb * NE + DT_RANK;      // row l -> + l*NE

#if HAVE_TDM
    const bool issuer = (tid < 32);                        // wave 0 drives the TDM
    if (issuer) tdm_load_tile32x32(bcsrc, &sBC[0][0][0]);
#else
    stage_bc_coop(bcsrc, &sBC[0][0][0], tid);
#endif

    int buf = 0;
    for (int l0 = 0; l0 < NL; l0 += LCHUNK) {
        const bool has_next = (l0 + LCHUNK) < NL;
        if (has_next) {
#if HAVE_TDM
            if (issuer)
                tdm_load_tile32x32(bcsrc + (size_t)(l0 + LCHUNK) * NE,
                                   &sBC[buf ^ 1][0][0]);
#else
            stage_bc_coop(bcsrc + (size_t)(l0 + LCHUNK) * NE,
                          &sBC[buf ^ 1][0][0], tid);
#endif
        }
#if HAVE_TDM
        if (issuer) { if (has_next) WAIT_TENSOR(1); else WAIT_TENSOR(0); }
#endif
        __syncthreads();                    // current buffer visible to all waves

        for (int li = 0; li < LCHUNK; ++li) {
            const size_t off = (rowb + l0 + li) * D_MODEL + d;
            const float dl = delta[off];
            const float ul = u[off];
            const float du = dl * ul;
            float accv = 0.f;
#pragma unroll
            for (int n = 0; n < D_STATE; ++n) {
                float dA = __expf(dl * a[n]);                     // v_exp_f32
                h[n] = fmaf(dA, h[n], du * sBC[buf][li][n]);      // h = dA*h + dBu
                accv = fmaf(h[n], sBC[buf][li][D_STATE + n], accv);
            }
            y[off] = fmaf(ul, dsk, accv);                         // + u*D_skip
        }
        __syncthreads();                    // all reads done before next overwrite
        buf ^= 1;
    }
}

// ---------------------------------------------------------------------------
// Host-side launcher
// ---------------------------------------------------------------------------
extern "C" void kernel_launch(void* const* d_in, const int* in_sizes, int n_in,
                              void* d_out, int out_size, void* d_ws, size_t ws_size,
                              hipStream_t stream)
{
    (void)in_sizes; (void)n_in; (void)out_size; (void)ws_size;
    const float* x    = (const float*)d_in[0];   // (B,L,D)
    const float* cw   = (const float*)d_in[1];   // (D,1,4)
    const float* cb   = (const float*)d_in[2];   // (D,)
    const float* Wx   = (const float*)d_in[3];   // (160,D)
    const float* Wdt  = (const float*)d_in[4];   // (D,128)
    const float* bdt  = (const float*)d_in[5];   // (D,)
    const float* Alog = (const float*)d_in[6];   // (D,16)
    const float* Dsk  = (const float*)d_in[7];   // (D,)
    float* y = (float*)d_out;                    // (B,L,D)

    char* ws = (char*)d_ws;
    const size_t u_bytes    = (size_t)NB * NL * D_MODEL * sizeof(float);
    const size_t xdbl_bytes = (((size_t)NB * NL * NE * sizeof(float)) + 255) & ~(size_t)255;
    float* u_ws     = (float*)(ws);
    float* xdbl_ws  = (float*)(ws + u_bytes);
    float* delta_ws = (float*)(ws + u_bytes + xdbl_bytes);

    conv_kernel<<<(NB * NL * D_MODEL) / 256, 256, 0, stream>>>(x, cw, cb, u_ws);
    xproj_kernel<<<((NB * NL / 16) * (NE / 16)) / 8, 256, 0, stream>>>(u_ws, Wx, xdbl_ws);
    dtproj_kernel<<<((NB * NL / 16) * (D_MODEL / 16)) / 8, 256, 0, stream>>>(
        xdbl_ws, Wdt, bdt, delta_ws);
    scan_kernel<<<(NB * D_MODEL) / 256, 256, 0, stream>>>(
        u_ws, xdbl_ws, delta_ws, Alog, Dsk, y);
}